// Network_34832184770685
// MI455X (gfx1250) — compile-verified
//
#include <hip/hip_runtime.h>

// Problem constants (from reference): 7 ops, B*C = 128*64 = 8192 SPD matrices of 30x30.
#define NOPS 7
#define BC   8192
#define NDIM 30
#define NP   32     // padded matrix dim (identity padding keeps all steps consistent)
#define LS   33     // LDS row stride in floats (33 -> bank-conflict free)
#define WPB  2      // waves per block (LDS budget: 2 * 7 * 32*33*4 = 59136 B)
#define KARCHER_ITERS 3
#define NS_OUTER 8  // Newton-Schulz iters for sqrt/invsqrt of X
#define NS_INNER 6  // Newton-Schulz iters for the per-op quarter roots

typedef __attribute__((ext_vector_type(2))) float v2f;
typedef __attribute__((ext_vector_type(8))) float v8f;

// ---------------------------------------------------------------------------
// Wave-cooperative 32x32 f32 matmul: C = A * B, all operands in LDS.
// Uses V_WMMA_F32_16X16X4_F32: 2x2 tiles of 16x16, K swept in chunks of 4.
// A-frag (16x4): lanes 0-15 row M=l hold K=k0,k0+1; lanes 16-31 hold K=k0+2,k0+3.
// B-frag (4x16): VGPR v holds rows {k0+v, k0+v+2} striped across lane halves.
// C-frag (16x16): VGPR r = rows r (lanes 0-15) and r+8 (lanes 16-31).
// Accumulation lives entirely in registers; LDS write-back happens only at the
// end, so in-place C==A or C==B is safe (LDS is in-order within a wave).
// ---------------------------------------------------------------------------
__device__ __forceinline__ void mm32(const float* A, const float* B, float* C,
                                     int lane) {
  const int l = lane & 15;
  const int h = lane >> 4;      // 0 for lanes 0-15, 1 for lanes 16-31
  v8f acc[2][2];
  acc[0][0] = (v8f){}; acc[0][1] = (v8f){};
  acc[1][0] = (v8f){}; acc[1][1] = (v8f){};

#pragma unroll
  for (int k = 0; k < 8; ++k) {            // K chunks of 4
    const int k0 = k * 4 + h * 2;
    v2f af[2], bf[2];
#pragma unroll
    for (int t = 0; t < 2; ++t) {
      const int row = t * 16 + l;          // A tile row
      af[t][0] = A[row * LS + k0];
      af[t][1] = A[row * LS + k0 + 1];
      const int col = t * 16 + l;          // B tile col
      bf[t][0] = B[k0 * LS + col];
      bf[t][1] = B[(k0 + 1) * LS + col];
    }
#pragma unroll
    for (int i = 0; i < 2; ++i)
#pragma unroll
      for (int j = 0; j < 2; ++j)
        acc[i][j] = __builtin_amdgcn_wmma_f32_16x16x4_f32(
            false, af[i], false, bf[j], (short)0, acc[i][j], false, false);
  }

#pragma unroll
  for (int i = 0; i < 2; ++i)
#pragma unroll
    for (int j = 0; j < 2; ++j)
#pragma unroll
      for (int r = 0; r < 8; ++r) {
        const int row = i * 16 + h * 8 + r;
        const int col = j * 16 + l;
        C[row * LS + col] = acc[i][j][r];
      }
}

// Frobenius norm of the padded 32x32 matrix, broadcast to all lanes.
__device__ __forceinline__ float wave_frob(const float* M, int lane) {
  float s = 0.f;
  for (int i = lane; i < NP * NP; i += 32) {
    const int r = i >> 5, c = i & 31;
    const float v = M[r * LS + c];
    s += v * v;
  }
#pragma unroll
  for (int off = 16; off > 0; off >>= 1) s += __shfl_xor(s, off, 32);
  return sqrtf(s);
}

// In-place symmetrization: each unordered pair handled by exactly one
// lane-iteration (upper triangle owner), so no intra-wave write/read races.
__device__ __forceinline__ void symmetrize(float* M, int lane) {
  for (int i = lane; i < NP * NP; i += 32) {
    const int r = i >> 5, c = i & 31;
    if (r <= c) {
      const float a = M[r * LS + c];
      const float b = M[c * LS + r];
      const float s = 0.5f * (a + b);
      M[r * LS + c] = s;
      M[c * LS + r] = s;
    }
  }
}

// Coupled Newton-Schulz: Y -> sqrt(X), Z -> invsqrt(X). T is scratch.
__device__ __forceinline__ void ns_sqrt_pair(const float* X, float* Y, float* Z,
                                             float* T, int lane, int iters) {
  const float c = wave_frob(X, lane);
  const float rc = 1.0f / c;
  for (int i = lane; i < NP * NP; i += 32) {
    const int r = i >> 5, cc = i & 31;
    Y[r * LS + cc] = X[r * LS + cc] * rc;
    Z[r * LS + cc] = (r == cc) ? 1.f : 0.f;
  }
  for (int it = 0; it < iters; ++it) {
    mm32(Z, Y, T, lane);                           // T = Z*Y
    for (int i = lane; i < NP * NP; i += 32) {     // T = 0.5*(3I - T)
      const int r = i >> 5, cc = i & 31;
      T[r * LS + cc] = 0.5f * (((r == cc) ? 3.f : 0.f) - T[r * LS + cc]);
    }
    mm32(Y, T, Y, lane);                           // Y = Y*T (in place)
    mm32(T, Z, Z, lane);                           // Z = T*Z (in place)
  }
  const float sc = sqrtf(c), rsc = 1.0f / sc;
  for (int i = lane; i < NP * NP; i += 32) {
    const int r = i >> 5, cc = i & 31;
    Y[r * LS + cc] *= sc;
    Z[r * LS + cc] *= rsc;
  }
}

// In-place Newton-Schulz square root of M (Ztmp, T are scratch).
__device__ __forceinline__ void ns_sqrt_inplace(float* M, float* Ztmp, float* T,
                                                int lane, int iters) {
  const float c = wave_frob(M, lane);
  const float rc = 1.0f / c;
  for (int i = lane; i < NP * NP; i += 32) {
    const int r = i >> 5, cc = i & 31;
    M[r * LS + cc] *= rc;
    Ztmp[r * LS + cc] = (r == cc) ? 1.f : 0.f;
  }
  for (int it = 0; it < iters; ++it) {
    mm32(Ztmp, M, T, lane);
    for (int i = lane; i < NP * NP; i += 32) {
      const int r = i >> 5, cc = i & 31;
      T[r * LS + cc] = 0.5f * (((r == cc) ? 3.f : 0.f) - T[r * LS + cc]);
    }
    mm32(M, T, M, lane);
    mm32(T, Ztmp, Ztmp, lane);
  }
  const float sc = sqrtf(c);
  for (int i = lane; i < NP * NP; i += 32) {
    const int r = i >> 5, cc = i & 31;
    M[r * LS + cc] *= sc;
  }
}

__global__ __launch_bounds__(32 * WPB) void karcher_spd_kernel(
    const float* __restrict__ spds, const float* __restrict__ alphas,
    float* __restrict__ out) {
  __shared__ float lds[WPB][7][NP * LS];
  const int wid = threadIdx.x >> 5;
  const int lane = threadIdx.x & 31;
  const int m = blockIdx.x * WPB + wid;   // grid exactly covers BC -> no guard

  float* bX  = &lds[wid][0][0];
  float* bY  = &lds[wid][1][0];
  float* bZ  = &lds[wid][2][0];
  float* bL  = &lds[wid][3][0];
  float* bT0 = &lds[wid][4][0];
  float* bT1 = &lds[wid][5][0];
  float* bT2 = &lds[wid][6][0];

  // Softmax over the 7 architecture logits (redundant per lane; trivial cost).
  float w[NOPS];
  float mx = alphas[0];
#pragma unroll
  for (int o = 1; o < NOPS; ++o) mx = fmaxf(mx, alphas[o]);
  float sum = 0.f;
#pragma unroll
  for (int o = 0; o < NOPS; ++o) { w[o] = __expf(alphas[o] - mx); sum += w[o]; }
  const float rsum = 1.0f / sum;
#pragma unroll
  for (int o = 0; o < NOPS; ++o) w[o] *= rsum;

  // X init: weighted arithmetic mean, identity-padded to 32x32.
  for (int i = lane; i < NP * NP; i += 32) {
    const int r = i >> 5, c = i & 31;
    float v;
    if (r < NDIM && c < NDIM) {
      v = 0.f;
#pragma unroll
      for (int o = 0; o < NOPS; ++o)
        v += w[o] * spds[((size_t)o * BC + m) * (NDIM * NDIM) + r * NDIM + c];
    } else {
      v = (r == c) ? 1.f : 0.f;
    }
    bX[r * LS + c] = v;
  }

  for (int kit = 0; kit < KARCHER_ITERS; ++kit) {
    // Y = X^{1/2}, Z = X^{-1/2}
    ns_sqrt_pair(bX, bY, bZ, bT0, lane, NS_OUTER);

    // L = 0
    for (int i = lane; i < NP * NP; i += 32) {
      const int r = i >> 5, c = i & 31;
      bL[r * LS + c] = 0.f;
    }

    for (int o = 0; o < NOPS; ++o) {
      // Stage A_o (identity-padded) into LDS; prefetch next op's tile.
      const float* Ao = spds + ((size_t)o * BC + m) * (NDIM * NDIM);
      if (o + 1 < NOPS)
        __builtin_prefetch(spds + ((size_t)(o + 1) * BC + m) * (NDIM * NDIM) +
                               lane * 28, 0, 1);
      for (int i = lane; i < NP * NP; i += 32) {
        const int r = i >> 5, c = i & 31;
        bT2[r * LS + c] = (r < NDIM && c < NDIM)
                              ? Ao[r * NDIM + c]
                              : ((r == c) ? 1.f : 0.f);
      }

      mm32(bZ, bT2, bT0, lane);           // T0 = Z * A
      mm32(bT0, bZ, bT1, lane);           // T1 = Z * A * Z  (mid)
      symmetrize(bT1, lane);

      // log(mid) via 2x NS sqrt (mid^{1/4}) + 3-term Mercator * 4
      ns_sqrt_inplace(bT1, bT0, bT2, lane, NS_INNER);
      ns_sqrt_inplace(bT1, bT0, bT2, lane, NS_INNER);
      for (int i = lane; i < NP * NP; i += 32) {      // E = R - I
        const int r = i >> 5, c = i & 31;
        if (r == c) bT1[r * LS + c] -= 1.f;
      }
      mm32(bT1, bT1, bT0, lane);          // E^2
      mm32(bT0, bT1, bT2, lane);          // E^3
      for (int i = lane; i < NP * NP; i += 32) {
        const int r = i >> 5, c = i & 31;
        const int idx = r * LS + c;
        bL[idx] += w[o] * 4.f *
                   (bT1[idx] - 0.5f * bT0[idx] + (1.f / 3.f) * bT2[idx]);
      }
    }

    symmetrize(bL, lane);

    // E = exp(L): scale by 1/16, 6-term Horner Taylor, square 4x.
    for (int i = lane; i < NP * NP; i += 32) {
      const int r = i >> 5, c = i & 31;
      bL[r * LS + c] *= (1.f / 16.f);
    }
    for (int i = lane; i < NP * NP; i += 32) {        // acc = I + L/6
      const int r = i >> 5, c = i & 31;
      bT0[r * LS + c] = ((r == c) ? 1.f : 0.f) + bL[r * LS + c] * (1.f / 6.f);
    }
    for (int k = 5; k >= 1; --k) {
      mm32(bL, bT0, bT1, lane);                       // T1 = L * acc
      const float rk = 1.f / (float)k;
      for (int i = lane; i < NP * NP; i += 32) {
        const int r = i >> 5, c = i & 31;
        bT0[r * LS + c] = ((r == c) ? 1.f : 0.f) + bT1[r * LS + c] * rk;
      }
    }
    for (int s = 0; s < 4; ++s) mm32(bT0, bT0, bT0, lane);  // squaring

    // X = Y * E * Y
    mm32(bY, bT0, bT1, lane);
    mm32(bT1, bY, bX, lane);
    symmetrize(bX, lane);
  }

  // Write the 30x30 result.
  for (int i = lane; i < NDIM * NDIM; i += 32) {
    const int r = i / NDIM, c = i % NDIM;
    out[(size_t)m * (NDIM * NDIM) + i] = bX[r * LS + c];
  }
}

extern "C" void kernel_launch(void* const* d_in, const int* in_sizes, int n_in,
                              void* d_out, int out_size, void* d_ws,
                              size_t ws_size, hipStream_t stream) {
  const float* spds = (const float*)d_in[0];     // [7, 128, 64, 30, 30] f32
  const float* alphas = (const float*)d_in[1];   // [7] f32
  float* out = (float*)d_out;                    // [128, 64, 30, 30] f32

  (void)in_sizes; (void)n_in; (void)out_size; (void)d_ws; (void)ws_size;

  dim3 grid(BC / WPB);      // 4096 blocks
  dim3 block(32 * WPB);     // 2 waves (wave32) per block
  karcher_spd_kernel<<<grid, block, 0, stream>>>(spds, alphas, out);
}